// RWKVAttention_87213605912998
// MI455X (gfx1250) — compile-verified
//
#include <hip/hip_runtime.h>
#include <math.h>

// Problem constants (from reference)
#define HIDDEN   2048
#define NUM_HEADS 16
#define HEAD_DIM 128
#define OUTD     2048
#define BB       4
#define TT       2048
#define MM       (BB*TT)     // 8192 rows (b*T+t)
#define KK       2048        // contraction dim (both GEMM flavors)
#define NN       2048        // output dim (both GEMM flavors)
#define NCHUNK   16
#define CHUNKLEN 128         // TT / NCHUNK

typedef float v2f __attribute__((ext_vector_type(2)));
typedef float v8f __attribute__((ext_vector_type(8)));

// LDS geometry (64 banks):
//  A: row-major [128][APITCH], APITCH=20 floats -> 80B rows (float4-aligned),
//     fragment b64 reads hit 16 distinct bank-pairs (20*l mod 64 distinct, l=0..15).
//  B: K-pair interleaved [KB/2][BP2] with element pairs contiguous:
//     addr(k,n) = (k>>1)*BP2 + n*2 + (k&1).  BP2=288 (mod 64 == 32) makes the
//     two lane-halves of a fragment b64 read occupy disjoint bank ranges
//     (banks C..C+31 vs C+32..C+63) -> conflict-free, single ds_load_b64/frag.
#define APITCH 20
#define BP2    288
#define KB     16
#define NKB    (KK/KB)       // 128 k-blocks

// ---------------------------------------------------------------------------
// fp32 GEMM on the WMMA pipe:  Out[m,n] = act( sum_k X'[m,k] * W[n,k] )
// X' = X (mix==null)  or  X[m]*mix + X[m-1]*(1-mix)  (RWKV token shift fused)
// act = sigmoid when do_sigmoid != 0.
// Block tile 128x128, 8 waves, wave tile 32x64 (2x4 of 16x16 WMMA tiles).
// Double-buffered LDS + register prefetch: one barrier per k-block.
// ---------------------------------------------------------------------------
__global__ __launch_bounds__(256) void gemm_wmma_f32(
    const float* __restrict__ X, const float* __restrict__ W,
    float* __restrict__ Out, const float* __restrict__ mix, int do_sigmoid)
{
  __shared__ __align__(16) float As[2][128 * APITCH];   // 2 x 10240 B
  __shared__ __align__(16) float Bs[2][(KB/2) * BP2];   // 2 x  9216 B

  const int tid    = threadIdx.x;
  const int wid    = tid >> 5;
  const int lane   = tid & 31;
  const int half   = lane >> 4;        // lane-half selects K-pair / M-offset
  const int l16    = lane & 15;
  const int wave_m = (wid & 3) * 32;   // 4 waves along M
  const int wave_n = (wid >> 2) * 64;  // 2 waves along N
  const int m0     = blockIdx.y * 128;
  const int n0     = blockIdx.x * 128;

  // per-thread staging slots: 2 x float4 for A/W (+ shift/mix operands)
  const int row_[2] = { (tid + 0)   >> 2, (tid + 256) >> 2 };
  const int c4_[2]  = { (tid + 0)   & 3,  (tid + 256) & 3  };

  float4 acur[2], aprev[2], amx[2], wv4[2];

  v8f acc[2][4];
#pragma unroll
  for (int mi = 0; mi < 2; ++mi)
#pragma unroll
    for (int ni = 0; ni < 4; ++ni)
#pragma unroll
      for (int j = 0; j < 8; ++j) acc[mi][ni][j] = 0.0f;

  // ---- global -> registers for k-block kb ----
  auto loadG = [&](int kb) {
#pragma unroll
    for (int s = 0; s < 2; ++s) {
      const int row = row_[s];
      const int gk  = kb * KB + c4_[s] * 4;
      const int gm  = m0 + row;
      acur[s] = *(const float4*)(X + (size_t)gm * KK + gk);
      if (mix != nullptr) {
        const int t = gm & (TT - 1);
        aprev[s] = make_float4(0.f, 0.f, 0.f, 0.f);
        if (t > 0) aprev[s] = *(const float4*)(X + (size_t)(gm - 1) * KK + gk);
        amx[s] = *(const float4*)(mix + gk);
      }
      wv4[s] = *(const float4*)(W + (size_t)(n0 + row) * KK + gk);
    }
  };

  // ---- registers -> LDS buffer ----
  auto storeL = [&](int buf) {
#pragma unroll
    for (int s = 0; s < 2; ++s) {
      const int row = row_[s];
      const int c4  = c4_[s];
      float4 val;
      if (mix != nullptr) {
        val.x = acur[s].x * amx[s].x + aprev[s].x * (1.0f - amx[s].x);
        val.y = acur[s].y * amx[s].y + aprev[s].y * (1.0f - amx[s].y);
        val.z = acur[s].z * amx[s].z + aprev[s].z * (1.0f - amx[s].z);
        val.w = acur[s].w * amx[s].w + aprev[s].w * (1.0f - amx[s].w);
      } else {
        val = acur[s];
      }
      *(float4*)(&As[buf][row * APITCH + c4 * 4]) = val;
      // B transpose: k = c4*4 + i -> pair (2*c4 + (i>>1)), element (i&1)
      *(float2*)(&Bs[buf][(c4 * 2 + 0) * BP2 + row * 2]) = make_float2(wv4[s].x, wv4[s].y);
      *(float2*)(&Bs[buf][(c4 * 2 + 1) * BP2 + row * 2]) = make_float2(wv4[s].z, wv4[s].w);
    }
  };

  // ---- compute one k-block from LDS buffer ----
  auto compute = [&](int buf) {
#pragma unroll
    for (int kk = 0; kk < KB; kk += 4) {
      v2f a[2], b[4];
#pragma unroll
      for (int mi = 0; mi < 2; ++mi)
        a[mi] = *(const v2f*)(&As[buf][(wave_m + mi * 16 + l16) * APITCH + kk + 2 * half]);
#pragma unroll
      for (int ni = 0; ni < 4; ++ni)
        b[ni] = *(const v2f*)(&Bs[buf][(kk / 2 + half) * BP2 + (wave_n + ni * 16 + l16) * 2]);
#pragma unroll
      for (int mi = 0; mi < 2; ++mi)
#pragma unroll
        for (int ni = 0; ni < 4; ++ni)
          acc[mi][ni] = __builtin_amdgcn_wmma_f32_16x16x4_f32(
              false, a[mi], false, b[ni], (short)0, acc[mi][ni], false, false);
    }
  };

  // ---- software pipeline ----
  loadG(0);
  storeL(0);
  __syncthreads();
  int buf = 0;
  for (int kb = 0; kb < NKB; ++kb) {
    if (kb + 1 < NKB) loadG(kb + 1);      // issue next block's global loads early
    compute(buf);                          // WMMA from current LDS buffer
    if (kb + 1 < NKB) storeL(buf ^ 1);     // write-ahead into alternate buffer
    __syncthreads();
    buf ^= 1;
  }

  // ---- epilogue: C/D layout VGPR j -> rows {j, j+8} split by lane-half ----
#pragma unroll
  for (int mi = 0; mi < 2; ++mi)
#pragma unroll
    for (int ni = 0; ni < 4; ++ni)
#pragma unroll
      for (int j = 0; j < 8; ++j) {
        const int gm = m0 + wave_m + mi * 16 + j + 8 * half;
        const int gn = n0 + wave_n + ni * 16 + l16;
        float v = acc[mi][ni][j];
        if (do_sigmoid) v = 1.0f / (1.0f + expf(-v));
        Out[(size_t)gm * NN + gn] = v;
      }
}

// ---------------------------------------------------------------------------
// WKV chunked linear scan:  s_t = w*s_{t-1} + k_t*v_t ;  out_t = (s_{t-1} + u*kv_t)*r_t
// Pass 1: per-(channel,chunk) decayed sum  S_c = sum_i w^(L-1-i) * kv_i
// ---------------------------------------------------------------------------
__global__ __launch_bounds__(256) void wkv_chunk_sum(
    const float* __restrict__ k, const float* __restrict__ v,
    const float* __restrict__ td, float* __restrict__ S)
{
  const int g  = blockIdx.x * 256 + threadIdx.x;  // B*H*NCHUNK*D = 131072
  const int d  = g & 127;
  const int c  = (g >> 7) & (NCHUNK - 1);
  const int bh = g >> 11;           // b*H + h
  const int h  = bh & (NUM_HEADS - 1);
  const int b  = bh >> 4;

  const float w = expf(-expf(td[h * HEAD_DIM + d]));
  const int base = b * TT * OUTD + h * HEAD_DIM + d;
  const int t0 = c * CHUNKLEN;
  float s = 0.0f;
  for (int i = 0; i < CHUNKLEN; ++i) {
    const int idx = base + (t0 + i) * OUTD;
    s = s * w + k[idx] * v[idx];
  }
  S[(bh * HEAD_DIM + d) * NCHUNK + c] = s;
}

// Pass 2: sequential combine over chunks; replace S_c with chunk ENTRY state E_c
__global__ __launch_bounds__(256) void wkv_combine(
    const float* __restrict__ td, float* __restrict__ S)
{
  const int g  = blockIdx.x * 256 + threadIdx.x;  // channel id, 8192 total
  const int d  = g & 127;
  const int h  = (g >> 7) & (NUM_HEADS - 1);
  const float wL = expf(-(float)CHUNKLEN * expf(td[h * HEAD_DIM + d]));  // w^L exactly
  float* Sp = S + (size_t)g * NCHUNK;
  float E = 0.0f;
  for (int c = 0; c < NCHUNK; ++c) {
    const float tmp = Sp[c];
    Sp[c] = E;                 // entry state for chunk c
    E = wL * E + tmp;
  }
}

// Pass 3: per-(channel,chunk) output pass; writes attn IN-PLACE over r
__global__ __launch_bounds__(256) void wkv_out(
    float* __restrict__ r, const float* __restrict__ k,
    const float* __restrict__ v, const float* __restrict__ td,
    const float* __restrict__ tf, const float* __restrict__ S)
{
  const int g  = blockIdx.x * 256 + threadIdx.x;
  const int d  = g & 127;
  const int c  = (g >> 7) & (NCHUNK - 1);
  const int bh = g >> 11;
  const int h  = bh & (NUM_HEADS - 1);
  const int b  = bh >> 4;
  const int hd = h * HEAD_DIM + d;

  const float w = expf(-expf(td[hd]));
  const float u = tf[hd];
  float state = S[(bh * HEAD_DIM + d) * NCHUNK + c];
  const int base = b * TT * OUTD + hd;
  const int t0 = c * CHUNKLEN;
  for (int i = 0; i < CHUNKLEN; ++i) {
    const int idx = base + (t0 + i) * OUTD;
    const float kv = k[idx] * v[idx];
    const float rr = r[idx];
    r[idx] = (state + u * kv) * rr;   // in-place: r becomes attn
    state = state * w + kv;
  }
}

// ---------------------------------------------------------------------------
extern "C" void kernel_launch(void* const* d_in, const int* in_sizes, int n_in,
                              void* d_out, int out_size, void* d_ws, size_t ws_size,
                              hipStream_t stream) {
  (void)in_sizes; (void)n_in; (void)out_size; (void)ws_size;
  const float* hidden = (const float*)d_in[0];
  const float* Wr     = (const float*)d_in[1];
  const float* Wk     = (const float*)d_in[2];
  const float* Wv     = (const float*)d_in[3];
  const float* Wo     = (const float*)d_in[4];
  const float* td     = (const float*)d_in[5];
  const float* tf     = (const float*)d_in[6];
  const float* mk     = (const float*)d_in[7];
  const float* mv     = (const float*)d_in[8];
  const float* mr     = (const float*)d_in[9];
  float* out = (float*)d_out;
  float* ws  = (float*)d_ws;

  const size_t SZ = (size_t)MM * OUTD;   // 16,777,216 floats per activation buffer
  float* rbuf = ws;                      // r -> later overwritten with attn
  float* kbuf = ws + SZ;
  float* vbuf = ws + 2 * SZ;
  float* Sbuf = ws + 3 * SZ;             // chunk states: 131072 floats

  const dim3 gg(NN / 128, MM / 128);     // 16 x 64 workgroups
  const dim3 bkt(256);

  // QKV projections (token-shift mixing + sigmoid fused)
  gemm_wmma_f32<<<gg, bkt, 0, stream>>>(hidden, Wr, rbuf, mr, 1);
  gemm_wmma_f32<<<gg, bkt, 0, stream>>>(hidden, Wk, kbuf, mk, 0);
  gemm_wmma_f32<<<gg, bkt, 0, stream>>>(hidden, Wv, vbuf, mv, 0);

  // chunked WKV scan (16x time-parallelism)
  wkv_chunk_sum<<<BB * NUM_HEADS * NCHUNK * HEAD_DIM / 256, 256, 0, stream>>>(kbuf, vbuf, td, Sbuf);
  wkv_combine  <<<BB * NUM_HEADS * HEAD_DIM / 256,          256, 0, stream>>>(td, Sbuf);
  wkv_out      <<<BB * NUM_HEADS * NCHUNK * HEAD_DIM / 256, 256, 0, stream>>>(rbuf, kbuf, vbuf, td, tf, Sbuf);

  // output projection: out[m,h'] = sum_o attn[m,o] * Wo[h',o]
  gemm_wmma_f32<<<gg, bkt, 0, stream>>>(rbuf, Wo, out, nullptr, 0);
}